// BlockSSM_84997402788530
// MI455X (gfx1250) — compile-verified
//
#include <hip/hip_runtime.h>

// ---------------------------------------------------------------------------
// BlockSSM forward on gfx1250 using fp32 WMMA (V_WMMA_F32_16X16X4_F32).
//   T=128 sequential steps; one kernel launch per step (129 total, the extra
//   one finishes Y[T-1]).  Step s computes xn/FU/FD/X[s] and Y[s-1] (reading
//   the fully-written X[s-1] from d_out, ordered by the stream).
// ---------------------------------------------------------------------------

typedef float v2f __attribute__((ext_vector_type(2)));
typedef float v8f __attribute__((ext_vector_type(8)));

#define T_STEPS 128
#define BSZ     512
#define NX      512
#define NU      128
#define ND      64
#define NY      128

#define NBLK_X  32   // (BSZ/128) * (NX/64)
#define NBLK_Y  8    // (BSZ/128) * (NY/64)

__device__ __forceinline__ v8f wmma_f32(v2f a, v2f b, v8f c) {
  // (neg_a, A, neg_b, B, c_mod, C, reuse_a, reuse_b)
  return __builtin_amdgcn_wmma_f32_16x16x4_f32(false, a, false, b, (short)0, c,
                                               false, false);
}

// D(16x64) += A(16xK, row-major lda) * W^T(Kx64), W row-major [cols x K].
// Wave-level: lane L holds A row (L&15), K-pair 2*(L>>4); B frag is the
// matching float2 from weight row (col).
template <int K>
__device__ __forceinline__ void gemm_tile(const float* __restrict__ A, int lda,
                                          int rowA,
                                          const float* __restrict__ W, int ldw,
                                          int colW, v8f acc[4], int lane) {
  const int r  = rowA + (lane & 15);
  const int kh = (lane >> 4) << 1;  // 0 or 2
  const float* arow = A + (size_t)r * lda + kh;
  const float* wrow = W + (size_t)(colW + (lane & 15)) * ldw + kh;
#pragma unroll 4
  for (int k = 0; k < K; k += 4) {
    v2f a = *(const v2f*)(arow + k);
#pragma unroll
    for (int nt = 0; nt < 4; ++nt) {
      v2f b = *(const v2f*)(wrow + (size_t)(nt * 16) * ldw + k);
      acc[nt] = wmma_f32(a, b, acc[nt]);
    }
  }
}

// C/D 16x16 f32 layout: vgpr v, lanes 0-15 -> M=v, lanes 16-31 -> M=v+8; N=lane&15.
__device__ __forceinline__ void store_tiles(float* __restrict__ out, int ldo,
                                            int rowBase, int colBase,
                                            const v8f acc[4], int lane) {
  const int n  = lane & 15;
  const int mo = (lane >> 4) << 3;  // 0 or 8
#pragma unroll
  for (int nt = 0; nt < 4; ++nt) {
    float* col = out + (size_t)(rowBase + mo) * ldo + colBase + nt * 16 + n;
#pragma unroll
    for (int v = 0; v < 8; ++v) col[(size_t)v * ldo] = acc[nt][v];
  }
}

__global__ void __launch_bounds__(256) ssm_step_kernel(
    const float* __restrict__ x0, const float* __restrict__ Uf,
    const float* __restrict__ Df, const float* __restrict__ Wx,
    const float* __restrict__ bx, const float* __restrict__ Wu,
    const float* __restrict__ bu, const float* __restrict__ Wd,
    const float* __restrict__ bd, const float* __restrict__ Wy,
    const float* __restrict__ by, float* __restrict__ out, int s) {
  const int lane = threadIdx.x & 31;
  const int wave = threadIdx.x >> 5;

  float* X   = out;                                  // [T,B,NX]
  float* Y   = X + (size_t)T_STEPS * BSZ * NX;       // [T,B,NY]
  float* FU  = Y + (size_t)T_STEPS * BSZ * NY;       // [T,B,NX]
  float* FD  = FU + (size_t)T_STEPS * BSZ * NX;      // [T,B,NX]
  float* REG = FD + (size_t)T_STEPS * BSZ * NX;      // [1]

  const int blk = blockIdx.x;

  if (blk < NBLK_X) {
    // -------- x-part: xn = 2*(x Wx^T + bx + fu) + fd; also FU, FD --------
    if (s >= T_STEPS) return;
    const int rowBase = (blk >> 3) * 128 + wave * 16;  // over B
    const int colBase = (blk & 7) * 64;                // over NX

    const float* xin = (s == 0) ? x0 : (X + (size_t)(s - 1) * BSZ * NX);
    const float* u   = Uf + (size_t)s * BSZ * NU;
    const float* d   = Df + (size_t)s * BSZ * ND;

    const int n    = lane & 15;
    float bu_l[4], bd_l[4], bx_l[4];
#pragma unroll
    for (int nt = 0; nt < 4; ++nt) {
      bu_l[nt] = bu[colBase + nt * 16 + n];
      bd_l[nt] = bd[colBase + nt * 16 + n];
      bx_l[nt] = bx[colBase + nt * 16 + n];
    }

    // fu = u Wu^T + bu
    v8f fu[4] = {v8f{}, v8f{}, v8f{}, v8f{}};
    gemm_tile<NU>(u, NU, rowBase, Wu, NU, colBase, fu, lane);
#pragma unroll
    for (int nt = 0; nt < 4; ++nt)
#pragma unroll
      for (int v = 0; v < 8; ++v) fu[nt][v] += bu_l[nt];
    store_tiles(FU + (size_t)s * BSZ * NX, NX, rowBase, colBase, fu, lane);

    // fd = d Wd^T + bd
    v8f fd[4] = {v8f{}, v8f{}, v8f{}, v8f{}};
    gemm_tile<ND>(d, ND, rowBase, Wd, ND, colBase, fd, lane);
#pragma unroll
    for (int nt = 0; nt < 4; ++nt)
#pragma unroll
      for (int v = 0; v < 8; ++v) fd[nt][v] += bd_l[nt];
    store_tiles(FD + (size_t)s * BSZ * NX, NX, rowBase, colBase, fd, lane);

    // acc = fu (C input), then acc += x Wx^T  ->  xn1 = acc + bx
    v8f acc[4] = {fu[0], fu[1], fu[2], fu[3]};
    gemm_tile<NX>(xin, NX, rowBase, Wx, NX, colBase, acc, lane);
#pragma unroll
    for (int nt = 0; nt < 4; ++nt)
#pragma unroll
      for (int v = 0; v < 8; ++v) {
        float xn1 = acc[nt][v] + bx_l[nt];
        acc[nt][v] = xn1 + (xn1 + fd[nt][v]);  // xn = 2*xn1 + fd
      }
    store_tiles(X + (size_t)s * BSZ * NX, NX, rowBase, colBase, acc, lane);

    if (s == 0 && blk == 0 && threadIdx.x == 0) REG[0] = 0.0f;  // reg_error
  } else {
    // -------- y-part: Y[s-1] = X[s-1] Wy^T + by --------
    if (s == 0) return;
    const int yblk    = blk - NBLK_X;
    const int rowBase = (yblk >> 1) * 128 + wave * 16;  // over B
    const int colBase = (yblk & 1) * 64;                // over NY

    const float* xprev = X + (size_t)(s - 1) * BSZ * NX;
    const int n = lane & 15;
    float by_l[4];
#pragma unroll
    for (int nt = 0; nt < 4; ++nt) by_l[nt] = by[colBase + nt * 16 + n];

    v8f acc[4] = {v8f{}, v8f{}, v8f{}, v8f{}};
    gemm_tile<NX>(xprev, NX, rowBase, Wy, NX, colBase, acc, lane);
#pragma unroll
    for (int nt = 0; nt < 4; ++nt)
#pragma unroll
      for (int v = 0; v < 8; ++v) acc[nt][v] += by_l[nt];
    store_tiles(Y + (size_t)(s - 1) * BSZ * NY, NY, rowBase, colBase, acc, lane);
  }
}

extern "C" void kernel_launch(void* const* d_in, const int* in_sizes, int n_in,
                              void* d_out, int out_size, void* d_ws,
                              size_t ws_size, hipStream_t stream) {
  // setup_inputs order: Yf, x0, Uf, Df, Wx, bx, Wu, bu, Wd, bd, Wy, by
  const float* x0 = (const float*)d_in[1];
  const float* Uf = (const float*)d_in[2];
  const float* Df = (const float*)d_in[3];
  const float* Wx = (const float*)d_in[4];
  const float* bx = (const float*)d_in[5];
  const float* Wu = (const float*)d_in[6];
  const float* bu = (const float*)d_in[7];
  const float* Wd = (const float*)d_in[8];
  const float* bd = (const float*)d_in[9];
  const float* Wy = (const float*)d_in[10];
  const float* by = (const float*)d_in[11];
  float* out = (float*)d_out;

  // s = 0..T: step s computes X/FU/FD[s] (s<T) and Y[s-1] (s>=1).
  for (int s = 0; s <= T_STEPS; ++s) {
    ssm_step_kernel<<<dim3(NBLK_X + NBLK_Y), dim3(256), 0, stream>>>(
        x0, Uf, Df, Wx, bx, Wu, bu, Wd, bd, Wy, by, out, s);
  }
}